// FragmentEmbedder_498216206597
// MI455X (gfx1250) — compile-verified
//
#include <hip/hip_runtime.h>
#include <stdint.h>

#define BLOCK  256
#define DEMB   5
#define DSINE  40
#define ROWB   (DSINE * DEMB * 4)          // 800 bytes per gathered weight row
#define BLKOUTB (BLOCK * DEMB * 4)         // 5120 bytes of output per block

// WGP-scope prefetch (scope field omitted -> SCOPE_WGP = 0: pull into all
// cache levels). Safe here: every prefetched address is inside the mapped
// weight1 or bias1 table, so the implied non-speculative behavior cannot fault.
#define PF_WGP(addr, OFS) \
    asm volatile("global_prefetch_b8 %0, off offset:" OFS :: "v"(addr))

// freqs[i] = 1000^(-2(i+1)/10), pre-divided by 2*pi for v_sin_f32/v_cos_f32
// (hardware sin/cos take the angle in revolutions).
__device__ __constant__ float kFreq2Pi[10] = {
    0.251188643150958f    * 0.15915494309189535f,
    0.063095734448019f    * 0.15915494309189535f,
    0.015848931924611f    * 0.15915494309189535f,
    0.003981071705535f    * 0.15915494309189535f,
    0.001f                * 0.15915494309189535f,
    2.51188643150958e-4f  * 0.15915494309189535f,
    6.3095734448019e-5f   * 0.15915494309189535f,
    1.5848931924611e-5f   * 0.15915494309189535f,
    3.981071705535e-6f    * 0.15915494309189535f,
    1.0e-6f               * 0.15915494309189535f,
};

__global__ __launch_bounds__(BLOCK)
void FragmentEmbedder_kernel(const float* __restrict__ coords,
                             const int*   __restrict__ gene_ix,
                             const float* __restrict__ weight1,
                             const float* __restrict__ bias1,
                             float*       __restrict__ out,
                             int N)
{
    __shared__ float smem[BLOCK * DEMB];   // 5 KB output staging, block-contiguous

    const int tid = threadIdx.x;
    const int n   = blockIdx.x * BLOCK + tid;

    if (n < N) {
        // Coordinate load first: independent of the gene_ix -> address ->
        // weight-row dependency chain, so it can overlap.
        const float2 c = ((const float2*)coords)[n];

        const int g = gene_ix[n];
        const float* __restrict__ wrow = weight1 + (size_t)g * (DSINE * DEMB);
        const float* __restrict__ brow = bias1   + (size_t)g * DEMB;

        // WGP-scope global_prefetch_b8: pull the gathered 800B row (table is
        // 4 MB -> GL2-resident) into the WGP cache while we burn ~40
        // transcendental ops below. 128B line stride + last-byte cover.
        {
            const char* wb = (const char*)wrow;
            PF_WGP(wb, "0");
            PF_WGP(wb, "128");
            PF_WGP(wb, "256");
            PF_WGP(wb, "384");
            PF_WGP(wb, "512");
            PF_WGP(wb, "640");
            PF_WGP(wb, "768");
            PF_WGP(wb, "799");            // last line if row is 128B-misaligned
            PF_WGP((const char*)brow, "0");
        }

        // emb[d*20 + 2i] = sin(c_d * f_i), emb[d*20 + 2i + 1] = cos(c_d * f_i)
        float emb[DSINE];
        #pragma unroll
        for (int i = 0; i < 10; ++i) {
            const float ax = c.x * kFreq2Pi[i];
            const float ay = c.y * kFreq2Pi[i];
            emb[2*i + 0]  = __builtin_amdgcn_sinf(ax);
            emb[2*i + 1]  = __builtin_amdgcn_cosf(ax);
            emb[20 + 2*i + 0] = __builtin_amdgcn_sinf(ay);
            emb[20 + 2*i + 1] = __builtin_amdgcn_cosf(ay);
        }

        float a0 = brow[0], a1 = brow[1], a2 = brow[2], a3 = brow[3], a4 = brow[4];

        // Row is 800B -> 16B aligned; stream it as 50 x b128 loads.
        const float4* __restrict__ wv = (const float4*)wrow;
        #pragma unroll
        for (int jj = 0; jj < 10; ++jj) {
            const float4 q0 = wv[jj*5 + 0];
            const float4 q1 = wv[jj*5 + 1];
            const float4 q2 = wv[jj*5 + 2];
            const float4 q3 = wv[jj*5 + 3];
            const float4 q4 = wv[jj*5 + 4];
            const float e0 = emb[4*jj + 0];
            const float e1 = emb[4*jj + 1];
            const float e2 = emb[4*jj + 2];
            const float e3 = emb[4*jj + 3];
            a0 = fmaf(e0, q0.x, a0);
            a1 = fmaf(e0, q0.y, a1);
            a2 = fmaf(e0, q0.z, a2);
            a3 = fmaf(e0, q0.w, a3);
            a4 = fmaf(e0, q1.x, a4);
            a0 = fmaf(e1, q1.y, a0);
            a1 = fmaf(e1, q1.z, a1);
            a2 = fmaf(e1, q1.w, a2);
            a3 = fmaf(e1, q2.x, a3);
            a4 = fmaf(e1, q2.y, a4);
            a0 = fmaf(e2, q2.z, a0);
            a1 = fmaf(e2, q2.w, a1);
            a2 = fmaf(e2, q3.x, a2);
            a3 = fmaf(e2, q3.y, a3);
            a4 = fmaf(e2, q3.z, a4);
            a0 = fmaf(e3, q3.w, a0);
            a1 = fmaf(e3, q4.x, a1);
            a2 = fmaf(e3, q4.y, a2);
            a3 = fmaf(e3, q4.z, a3);
            a4 = fmaf(e3, q4.w, a4);
        }

        // sigmoid(x) = rcp(1 + exp2(-x * log2(e)))  -> v_exp_f32 + v_rcp_f32
        const float L2E = 1.4426950408889634f;
        a0 = __builtin_amdgcn_rcpf(1.0f + __builtin_amdgcn_exp2f(-a0 * L2E));
        a1 = __builtin_amdgcn_rcpf(1.0f + __builtin_amdgcn_exp2f(-a1 * L2E));
        a2 = __builtin_amdgcn_rcpf(1.0f + __builtin_amdgcn_exp2f(-a2 * L2E));
        a3 = __builtin_amdgcn_rcpf(1.0f + __builtin_amdgcn_exp2f(-a3 * L2E));
        a4 = __builtin_amdgcn_rcpf(1.0f + __builtin_amdgcn_exp2f(-a4 * L2E));

        smem[tid*5 + 0] = a0;
        smem[tid*5 + 1] = a1;
        smem[tid*5 + 2] = a2;
        smem[tid*5 + 3] = a3;
        smem[tid*5 + 4] = a4;
    }

    __syncthreads();

    // Coalesced drain: LDS -> global via gfx1250 async store (ASYNCcnt path).
    // Block's output region is 16B aligned (blockIdx * 5120B).
    const long long baseByte  = (long long)blockIdx.x * BLKOUTB;
    const long long totalByte = (long long)N * (DEMB * 4);
    long long rem = totalByte - baseByte;
    const int lim   = (rem < (long long)BLKOUTB) ? (int)rem : BLKOUTB;
    const int lim16 = lim & ~15;

    char* outb = (char*)out + baseByte;
    const uint32_t ldsBase = (uint32_t)(uintptr_t)&smem[0];  // low 32 bits of flat = LDS addr

    #pragma unroll
    for (int chunk = 0; chunk < 2; ++chunk) {
        const int b = chunk * (BLOCK * 16) + tid * 16;
        if (b < lim16) {
            const char*    gaddr = outb + b;
            const uint32_t laddr = ldsBase + (uint32_t)b;
            asm volatile("global_store_async_from_lds_b128 %0, %1, off"
                         :: "v"(gaddr), "v"(laddr)
                         : "memory");
        }
    }
    // Sub-16B remainder (empty for N = 1e6, kept for generality): dword stores.
    const int tb = lim16 + tid * 4;
    if (tb < lim) {
        *(float*)(outb + tb) = smem[tb >> 2];
    }

    asm volatile("s_wait_asynccnt 0" ::: "memory");
}

extern "C" void kernel_launch(void* const* d_in, const int* in_sizes, int n_in,
                              void* d_out, int out_size, void* d_ws, size_t ws_size,
                              hipStream_t stream) {
    const float* coords  = (const float*)d_in[0];   // [N,2] f32
    const int*   gene_ix = (const int*)d_in[1];     // [N]   i32
    const float* weight1 = (const float*)d_in[2];   // [5000,40,5] f32
    const float* bias1   = (const float*)d_in[3];   // [5000,5] f32
    float*       out     = (float*)d_out;           // [N,5] f32

    const int N = in_sizes[1];
    const int blocks = (N + BLOCK - 1) / BLOCK;
    FragmentEmbedder_kernel<<<blocks, BLOCK, 0, stream>>>(
        coords, gene_ix, weight1, bias1, out, N);
}